// kNN_6622839570654
// MI455X (gfx1250) — compile-verified
//
#include <hip/hip_runtime.h>
#include <hip/hip_bf16.h>

// 1-NN classifier: B=2048 queries, N=32768 supports, D=256, 100 classes.
// cross = x @ support^T on V_WMMA_F32_16X16X4_F32 (fp32 matrix pipe).
// Support tiles are DMA'd into a double-buffered LDS region by the Tensor
// Data Mover (tensor_load_to_lds, TENSORcnt) so DMA of tile it+1 overlaps
// the WMMA chain on tile it. argmin over n of (||s||^2 - 2*cross).

#define BQ 2048
#define NS 32768
#define DD 256
#define NC 100
#define NSPLIT 4               // grid.y N-splits, combined via packed u64 atomicMin
#define STAGE_ROWS 64          // support rows per LDS tile
#define ROW_W 260              // padded LDS row stride (floats): 4-DW pad / 256 DW
                               // -> read banks (4*row + 2*half) % 64: all 64 banks once
#define BUF_FLOATS (STAGE_ROWS * ROW_W)   // 16640 floats = 66,560 B per buffer

typedef float v2f __attribute__((ext_vector_type(2)));
typedef float v8f __attribute__((ext_vector_type(8)));
typedef float v4f __attribute__((ext_vector_type(4)));

#if __has_builtin(__builtin_amdgcn_tensor_load_to_lds) && \
    __has_builtin(__builtin_amdgcn_s_wait_tensorcnt)
#define USE_TDM 1
typedef unsigned v4u_t __attribute__((ext_vector_type(4)));
typedef int      v8i_t __attribute__((ext_vector_type(8)));
typedef int      v4i_t __attribute__((ext_vector_type(4)));
#else
#define USE_TDM 0
#endif

__device__ __forceinline__ unsigned f2ord(float f) {
    unsigned u = __float_as_uint(f);
    return (u & 0x80000000u) ? ~u : (u | 0x80000000u);  // order-preserving map
}

#if USE_TDM
// Issue one TDM load: 64 rows x 256 f32 tile (row stride 256 elems in memory)
// into LDS at byte offset ldsOff, padded +16B per 1KB row (D# pad fields).
// This toolchain's builtin takes 6 args:
//   (uint32x4 group0, int32x8 group1, int32x4, int32x4, int32x8, i32 cpol)
__device__ __forceinline__ void tdm_load_tile(const float* gsrc, unsigned ldsOff) {
    unsigned long long ga = (unsigned long long)(size_t)gsrc;
    // D# group0: count=1 | lds_addr | global_addr[56:0] | type=2 (bits 127:126)
    v4u_t g0 = { 1u,
                 ldsOff,
                 (unsigned)ga,
                 (unsigned)((ga >> 32) & 0x01FFFFFFull) | 0x80000000u };
    // D# group1:
    //  d0: data_size=2(4B)<<16 | pad_enable<<20 | pad_interval=7(256DW)<<22
    //      | pad_amount=3(4DW)<<25
    //  tensor_dim0=256 (bits 79:48), tensor_dim1=32768 (bits 111:80),
    //  tile_dim0=256 (127:112), tile_dim1=64 (143:128), tile_dim2=0,
    //  tensor_dim0_stride=256 (207:160), dim1_stride=0 (2D tile).
    v8i_t g1 = { (int)((2u << 16) | (1u << 20) | (7u << 22) | (3u << 25)),
                 (int)(256u << 16),        // tensor_dim0[15:0] << 16
                 (int)(32768u << 16),      // tensor_dim0[31:16]=0 | tensor_dim1[15:0]<<16
                 (int)(256u << 16),        // tensor_dim1[31:16]=0 | tile_dim0<<16
                 64,                       // tile_dim1 | tile_dim2=0
                 256,                      // tensor_dim0_stride[31:0]
                 0, 0 };
    v4i_t z4 = { 0, 0, 0, 0 };             // groups 2/3 unused (2D tensor)
    v8i_t z8 = { 0, 0, 0, 0, 0, 0, 0, 0 }; // unused trailing group (count=1)
    __builtin_amdgcn_tensor_load_to_lds(g0, g1, z4, z4, z8, 0);
}
#endif

// ---------------------------------------------------------------------------
// Kernel 1: s2[n] = ||support[n]||^2 (wave-per-row, coalesced) + init packed.
// ---------------------------------------------------------------------------
__global__ __launch_bounds__(256)
void knn_prep(const float* __restrict__ sup, float* __restrict__ s2,
              unsigned long long* __restrict__ packed) {
    int gtid = blockIdx.x * blockDim.x + threadIdx.x;
    int row  = gtid >> 5;
    int lane = gtid & 31;
    if (row < NS) {
        const v4f* r = (const v4f*)(sup + (size_t)row * DD);
        v4f a = r[lane];
        v4f b = r[lane + 32];
        float acc = a.x * a.x + a.y * a.y + a.z * a.z + a.w * a.w
                  + b.x * b.x + b.y * b.y + b.z * b.z + b.w * b.w;
        #pragma unroll
        for (int off = 16; off > 0; off >>= 1)
            acc += __shfl_down(acc, off, 32);
        if (lane == 0) s2[row] = acc;
    }
    if (gtid < BQ) packed[gtid] = 0xFFFFFFFFFFFFFFFFull;
}

// ---------------------------------------------------------------------------
// Kernel 2: main WMMA kernel. 128 threads (4 waves); 16 query rows per block,
// N/NSPLIT supports. A-fragments (16 rows x K=256) live in registers
// (64 x v2f / lane). Support tiles double-buffered in LDS via TDM.
// ---------------------------------------------------------------------------
__global__ __launch_bounds__(128)
void knn_main(const float* __restrict__ x, const float* __restrict__ sup,
              const float* __restrict__ s2, unsigned long long* __restrict__ packed) {
    extern __shared__ float smem[];                  // 2 * BUF_FLOATS floats

    const int lane = threadIdx.x & 31;
    const int wave = threadIdx.x >> 5;
    const int m0   = blockIdx.x * 16;
    const int nStart = blockIdx.y * (NS / NSPLIT);
    const int kb   = (lane >> 4) << 1;               // K sub-offset: 0 or 2
    const int mrow = m0 + (lane & 15);               // A: M = lane&15

    // Preload A fragments for full K=256 (ISA 16x4 f32 layout).
    v2f afrag[64];
    #pragma unroll
    for (int kk = 0; kk < 64; ++kk)
        afrag[kk] = *(const v2f*)(x + mrow * DD + (kk << 2) + kb);

    float bestv[8];
    int   besti[8];
    #pragma unroll
    for (int j = 0; j < 8; ++j) { bestv[j] = 3.4e38f; besti[j] = 0x7FFFFFFF; }

    const int iters = (NS / NSPLIT) / STAGE_ROWS;    // 128

#if USE_TDM
    // Flat address of LDS buffer: low 32 bits are the LDS byte offset.
    const unsigned smemBase = (unsigned)(size_t)(void*)smem;
    if (wave == 0)
        tdm_load_tile(sup + (size_t)nStart * DD, smemBase);   // prefetch tile 0
#endif

    for (int it = 0; it < iters; ++it) {
        const int rowBase = nStart + (it << 6);
        float* cur = smem + (it & 1) * BUF_FLOATS;

#if USE_TDM
        __syncthreads();   // all waves done reading the 'next' buffer (tile it-1)
        if (wave == 0) {
            if (it + 1 < iters) {
                tdm_load_tile(sup + (size_t)(rowBase + STAGE_ROWS) * DD,
                              smemBase + (unsigned)(((it + 1) & 1) * BUF_FLOATS * 4));
                __builtin_amdgcn_s_wait_tensorcnt(1);   // tile 'it' has landed
            } else {
                __builtin_amdgcn_s_wait_tensorcnt(0);
            }
        }
        __syncthreads();   // release: tile 'it' visible to all waves
#else
        __syncthreads();
        // Fallback: synchronous coalesced staging into padded rows.
        #pragma unroll
        for (int i = 0; i < 32; ++i) {
            int t = i * 128 + threadIdx.x;           // 0..4095
            int r = t >> 6;                          // staged row 0..63
            int c = t & 63;                          // 16B chunk 0..63
            v4f v = *(const v4f*)(sup + (size_t)(rowBase + r) * DD + (c << 2));
            *(v4f*)(cur + r * ROW_W + (c << 2)) = v;
        }
        __syncthreads();
#endif

        // Wave w computes 16x16 tile: cols rowBase + w*16 + (lane&15).
        const int rloc = (wave << 4) + (lane & 15);
        const float* lrow = cur + rloc * ROW_W + kb;
        v8f acc = {0.f, 0.f, 0.f, 0.f, 0.f, 0.f, 0.f, 0.f};
        #pragma unroll
        for (int kk = 0; kk < 64; ++kk) {
            v2f bf = *(const v2f*)(lrow + (kk << 2));
            acc = __builtin_amdgcn_wmma_f32_16x16x4_f32(
                    false, afrag[kk], false, bf, (short)0, acc, false, false);
        }

        const int n = rowBase + (wave << 4) + (lane & 15);
        const float s2n = s2[n];
        #pragma unroll
        for (int j = 0; j < 8; ++j) {
            float sc = fmaf(-2.0f, acc[j], s2n);     // dist^2 - q^2 (monotone)
            if (sc < bestv[j]) { bestv[j] = sc; besti[j] = n; }
        }
    }

    // Block reduction: C layout -> VGPR j holds row j (lanes 0-15) / j+8 (16-31).
    __syncthreads();
    float* redv = smem;                              // [4][8][32] floats
    int*   redi = (int*)(smem + 1024);               // [4][8][32] ints
    #pragma unroll
    for (int j = 0; j < 8; ++j) {
        redv[(wave * 8 + j) * 32 + lane] = bestv[j];
        redi[(wave * 8 + j) * 32 + lane] = besti[j];
    }
    __syncthreads();
    if (threadIdx.x < 16) {
        int r = threadIdx.x;                         // local query row 0..15
        int j = r & 7;
        int laneBase = (r >> 3) << 4;                // 0 or 16
        float bv = 3.4e38f; int bi = 0x7FFFFFFF;
        for (int w = 0; w < 4; ++w)
            for (int l = 0; l < 16; ++l) {
                float v = redv[(w * 8 + j) * 32 + laneBase + l];
                int   i = redi[(w * 8 + j) * 32 + laneBase + l];
                if (v < bv || (v == bv && i < bi)) { bv = v; bi = i; }
            }
        unsigned long long key = ((unsigned long long)f2ord(bv) << 32) | (unsigned)bi;
        atomicMin(&packed[m0 + r], key);             // GLOBAL_ATOMIC_MIN_U64
    }
}

// ---------------------------------------------------------------------------
// Kernel 3: unpack winner, gather int64 label, emit int32 one-hot.
// ---------------------------------------------------------------------------
__global__ __launch_bounds__(128)
void knn_onehot(const unsigned long long* __restrict__ packed,
                const long long* __restrict__ labels, int* __restrict__ out) {
    int r = blockIdx.x * blockDim.x + threadIdx.x;
    if (r >= BQ) return;
    int idx = (int)(unsigned)(packed[r] & 0xFFFFFFFFull);
    int lab = (int)labels[idx];
    #pragma unroll 4
    for (int c = 0; c < NC; ++c)
        out[r * NC + c] = (c == lab) ? 1 : 0;
}

extern "C" void kernel_launch(void* const* d_in, const int* in_sizes, int n_in,
                              void* d_out, int out_size, void* d_ws, size_t ws_size,
                              hipStream_t stream) {
    const float*     x    = (const float*)d_in[0];
    const float*     sup  = (const float*)d_in[1];
    const long long* labs = (const long long*)d_in[2];
    // d_in[3] = num_classes (scalar) -> fixed at 100 per reference
    int* out = (int*)d_out;

    float* s2 = (float*)d_ws;                                        // 128 KB
    unsigned long long* packed =
        (unsigned long long*)((char*)d_ws + NS * sizeof(float));     // 16 KB

    const int dynLds = 2 * BUF_FLOATS * (int)sizeof(float);          // 133,120 B
    // gfx1250 WGP has 320KB LDS; opt in to >64KB dynamic LDS (idempotent,
    // not a stream op, capture-safe).
    (void)hipFuncSetAttribute((const void*)knn_main,
                              hipFuncAttributeMaxDynamicSharedMemorySize, dynLds);

    knn_prep<<<(NS * 32) / 256, 256, 0, stream>>>(sup, s2, packed);
    knn_main<<<dim3(BQ / 16, NSPLIT), 128, dynLds, stream>>>(x, sup, s2, packed);
    knn_onehot<<<BQ / 128, 128, 0, stream>>>(packed, labs, out);
}